// PostProcess_13975823581337
// MI455X (gfx1250) — compile-verified
//
#include <hip/hip_runtime.h>
#include <stdint.h>

// ---------------- problem constants (match reference) ----------------
#define BATCH     32
#define NANCH     8732
#define NCLASS    81
#define NCLS      80          // classes excluding background
#define K_PRE     1000        // PRE_NMS_K
#define MAXOUT    100         // MAX_OUTPUT
#define CAND_MAX  32768       // per-batch candidate cap (expected ~18K valid)
#define NPAD      32768       // bitonic sort size (power of two, == CAND_MAX)

#define SCALE_XY_C  0.1f
#define SCALE_WH_C  0.2f
#define SCORE_THR_C 0.05f
#define CRITERIA_C  0.5f
#define MIN_SIZE_C  (0.1f / 300.0f)

// =====================================================================
// Kernel 1: box decode (+clip), online softmax over 81 classes,
// threshold test, push packed sort keys:
//   key = (float_bits(score) << 32) | ~idx     (idx = n*80 + (c-1))
// Descending sort on key == descending score, ascending idx tie-break,
// matching jax.lax.top_k on the masked (N*80) array.
// =====================================================================
__global__ void ssd_decode_softmax_kernel(
    const float* __restrict__ bboxes_in,   // (B, 4, N)
    const float* __restrict__ scores_in,   // (B, C, N)
    const float* __restrict__ dboxes,      // (N, 4) cx,cy,w,h
    float4* __restrict__ boxes_dec,        // (B, N) clipped ltrb
    int* __restrict__ cand_cnt,            // (B)
    unsigned long long* __restrict__ cand_keys) // (B, CAND_MAX)
{
    int g = blockIdx.x * blockDim.x + threadIdx.x;
    if (g >= BATCH * NANCH) return;
    int b = g / NANCH;
    int n = g - b * NANCH;

    // ---- decode one box ----
    const float* bb = bboxes_in + (size_t)b * 4 * NANCH + n;
    float bx = bb[0];
    float by = bb[(size_t)1 * NANCH];
    float bw = bb[(size_t)2 * NANCH];
    float bh = bb[(size_t)3 * NANCH];
    float4 d = ((const float4*)dboxes)[n];       // cx, cy, w, h

    float x = bx * SCALE_XY_C * d.z + d.x;
    float y = by * SCALE_XY_C * d.w + d.y;
    float w = __expf(bw * SCALE_WH_C) * d.z;
    float h = __expf(bh * SCALE_WH_C) * d.w;

    float l  = fminf(fmaxf(x - 0.5f * w, 0.f), 1.f);
    float t  = fminf(fmaxf(y - 0.5f * h, 0.f), 1.f);
    float r  = fminf(fmaxf(x + 0.5f * w, 0.f), 1.f);
    float bo = fminf(fmaxf(y + 0.5f * h, 0.f), 1.f);
    boxes_dec[g] = make_float4(l, t, r, bo);
    bool size_ok = ((r - l) >= MIN_SIZE_C) && ((bo - t) >= MIN_SIZE_C);

    // ---- online softmax over 81 classes (coalesced across n) ----
    const float* srow = scores_in + (size_t)b * NCLASS * NANCH + n;
    float m = -3.0e38f, s = 0.f;
    for (int c = 0; c < NCLASS; ++c) {
        if (c + 8 < NCLASS)
            __builtin_prefetch(&srow[(size_t)(c + 8) * NANCH], 0, 0); // global_prefetch_b8
        float v = srow[(size_t)c * NANCH];
        float mn = fmaxf(m, v);
        s = s * __expf(m - mn) + __expf(v - mn);
        m = mn;
    }
    float inv_s = 1.f / s;

    // ---- threshold + candidate push (classes 1..80) ----
    if (size_ok) {
        for (int c = 1; c < NCLASS; ++c) {
            float v = srow[(size_t)c * NANCH];       // L2-resident second pass
            float p = __expf(v - m) * inv_s;
            if (p > SCORE_THR_C) {
                int slot = atomicAdd(&cand_cnt[b], 1);
                if (slot < CAND_MAX) {
                    unsigned idx = (unsigned)n * NCLS + (unsigned)(c - 1);
                    unsigned long long key =
                        ((unsigned long long)__float_as_uint(p) << 32) |
                        (unsigned long long)(~idx);
                    cand_keys[(size_t)b * CAND_MAX + slot] = key;
                }
            }
        }
    }
}

// =====================================================================
// Kernel 2: per-batch top-1000 via full LDS bitonic sort of 32768 u64
// keys (256 KB of the WGP's 320 KB LDS). Keys streamed in with
// CDNA5 async global->LDS copies, drained with s_wait_asynccnt.
// Padding key 0 sorts last (all valid score bits > bits(0.05) > 0).
// =====================================================================
__global__ void ssd_topk_sort_kernel(
    const unsigned long long* __restrict__ cand_keys,
    const int* __restrict__ cand_cnt,
    const float4* __restrict__ boxes_dec,
    float4* __restrict__ tk_box,    // (B, K_PRE)
    int* __restrict__ tk_label,     // (B, K_PRE)
    float* __restrict__ tk_score)   // (B, K_PRE)  (-1 for invalid)
{
    extern __shared__ unsigned long long sh[];   // NPAD u64 (dyn LDS base == 0)
    const int b   = blockIdx.x;
    const int tid = threadIdx.x;
    const int NT  = blockDim.x;

    int count = cand_cnt[b];
    if (count > CAND_MAX) count = CAND_MAX;

    const unsigned long long gbase =
        (unsigned long long)(uintptr_t)(cand_keys + (size_t)b * CAND_MAX);

    for (int i = tid; i < NPAD; i += NT) {
        if (i < count) {
            unsigned off = (unsigned)i * 8u;   // same byte offset in LDS & memory
            asm volatile("global_load_async_to_lds_b64 %0, %1, %2"
                         :: "v"(off), "v"(off), "s"(gbase) : "memory");
        } else {
            sh[i] = 0ULL;
        }
    }
    asm volatile("s_wait_asynccnt 0" ::: "memory");
    __syncthreads();

    // bitonic sort, descending
    for (int k = 2; k <= NPAD; k <<= 1) {
        for (int j = k >> 1; j > 0; j >>= 1) {
            for (int i = tid; i < NPAD; i += NT) {
                int ixj = i ^ j;
                if (ixj > i) {
                    unsigned long long a = sh[i], c = sh[ixj];
                    bool descBlock = ((i & k) == 0);
                    bool sw = descBlock ? (a < c) : (a > c);
                    if (sw) { sh[i] = c; sh[ixj] = a; }
                }
            }
            __syncthreads();
        }
    }

    // emit sorted top-1000
    for (int i = tid; i < K_PRE; i += NT) {
        unsigned long long key = sh[i];
        size_t o = (size_t)b * K_PRE + i;
        if (key != 0ULL) {
            unsigned idx = ~(unsigned)(key & 0xffffffffu);
            unsigned bi  = idx / NCLS;
            tk_box[o]   = boxes_dec[(size_t)b * NANCH + bi];
            tk_label[o] = (int)(idx % NCLS) + 1;
            tk_score[o] = __uint_as_float((unsigned)(key >> 32));
        } else {
            tk_box[o]   = make_float4(0.f, 0.f, 0.f, 0.f);
            tk_label[o] = 0;
            tk_score[o] = -1.0f;
        }
    }
}

// =====================================================================
// Kernel 3: bitmask NMS + final top-100.
// Parallel: 1000x1000 label-masked IoU -> suppression bit-matrix in LDS.
// Wave32-parallel recurrence (exact replay of reference fori_loop):
//   per row i: owner lane broadcasts keep bit via __shfl (ds_bpermute),
//   if set, all 32 lanes AND-NOT their keep word against sup row i.
// List is score-descending, so first 100 kept == top_k(final_sc, 100).
// =====================================================================
#define NMS_WORDS 32   // ceil(1000/32)

__global__ void ssd_nms_out_kernel(
    const float4* __restrict__ tk_box,
    const int* __restrict__ tk_label,
    const float* __restrict__ tk_score,
    float* __restrict__ out_boxes,   // (B, 100, 4)
    int* __restrict__ out_labels,    // (B, 100)
    float* __restrict__ out_scores)  // (B, 100)
{
    extern __shared__ unsigned char smem[];
    unsigned* sup = (unsigned*)smem;                               // K_PRE*32 u32
    float4* bx = (float4*)(smem + (size_t)K_PRE * NMS_WORDS * 4);  // K_PRE
    int*    lb = (int*)((unsigned char*)bx + (size_t)K_PRE * 16);  // K_PRE
    float*  sc = (float*)((unsigned char*)lb + (size_t)K_PRE * 4); // K_PRE
    unsigned* keepw = (unsigned*)((unsigned char*)sc + (size_t)K_PRE * 4); // 32

    const int b   = blockIdx.x;
    const int tid = threadIdx.x;
    const int NT  = blockDim.x;

    for (int i = tid; i < K_PRE; i += NT) {
        bx[i] = tk_box[(size_t)b * K_PRE + i];
        lb[i] = tk_label[(size_t)b * K_PRE + i];
        sc[i] = tk_score[(size_t)b * K_PRE + i];
    }
    __syncthreads();

    // suppression bit-matrix: sup[i][w] bit jj <=> j=w*32+jj suppressed by i
    for (int t = tid; t < K_PRE * NMS_WORDS; t += NT) {
        int i = t >> 5;
        int w = t & 31;
        float4 a = bx[i];
        int   li = lb[i];
        float areaA = (a.z - a.x) * (a.w - a.y);
        unsigned bits = 0;
        for (int jj = 0; jj < 32; ++jj) {
            int j = w * 32 + jj;
            if (j < K_PRE && j > i && lb[j] == li) {
                float4 c = bx[j];
                float iw = fmaxf(fminf(a.z, c.z) - fmaxf(a.x, c.x), 0.f);
                float ih = fmaxf(fminf(a.w, c.w) - fmaxf(a.y, c.y), 0.f);
                float inter = iw * ih;
                float areaB = (c.z - c.x) * (c.w - c.y);
                float uni = areaA + areaB - inter;
                float iou = inter / fmaxf(uni, 1e-12f);
                if (iou > CRITERIA_C) bits |= 1u << jj;
            }
        }
        sup[i * NMS_WORDS + w] = bits;
    }
    __syncthreads();

    // wave32-parallel suppression recurrence on wave 0 (no barriers needed)
    if (tid < NMS_WORDS) {
        unsigned kw = 0;
        for (int jj = 0; jj < 32; ++jj) {
            int j = tid * 32 + jj;
            if (j < K_PRE && sc[j] > -0.5f) kw |= 1u << jj;   // cvalid init
        }
        for (int i = 0; i < K_PRE; ++i) {
            unsigned ow = __shfl(kw, i >> 5, 32);             // owner lane's word
            if ((ow >> (i & 31)) & 1u) {                      // keep[i] still set?
                kw &= ~sup[i * NMS_WORDS + tid];              // parallel AND-NOT
            }
        }
        keepw[tid] = kw;
    }
    __syncthreads();

    if (tid == 0) {
        // compact kept entries in ascending i == descending score order
        float* ob = out_boxes  + (size_t)b * MAXOUT * 4;
        int*   ol = out_labels + (size_t)b * MAXOUT;
        float* os = out_scores + (size_t)b * MAXOUT;
        int cnt = 0;
        for (int w = 0; w < NMS_WORDS && cnt < MAXOUT; ++w) {
            unsigned bits = keepw[w];
            while (bits != 0u && cnt < MAXOUT) {
                int j = __ffs(bits) - 1;
                bits &= bits - 1u;
                int i = w * 32 + j;
                float4 a = bx[i];
                ob[cnt * 4 + 0] = a.x; ob[cnt * 4 + 1] = a.y;
                ob[cnt * 4 + 2] = a.z; ob[cnt * 4 + 3] = a.w;
                ol[cnt] = lb[i];
                os[cnt] = sc[i];
                ++cnt;
            }
        }
        for (; cnt < MAXOUT; ++cnt) {
            ob[cnt * 4 + 0] = 0.f; ob[cnt * 4 + 1] = 0.f;
            ob[cnt * 4 + 2] = 0.f; ob[cnt * 4 + 3] = 0.f;
            ol[cnt] = 0;
            os[cnt] = 0.f;
        }
    }
}

// =====================================================================
extern "C" void kernel_launch(void* const* d_in, const int* in_sizes, int n_in,
                              void* d_out, int out_size, void* d_ws, size_t ws_size,
                              hipStream_t stream)
{
    const float* bboxes = (const float*)d_in[0];   // (32, 4, 8732)
    const float* scores = (const float*)d_in[1];   // (32, 81, 8732)
    const float* dboxes = (const float*)d_in[2];   // (8732, 4)

    unsigned char* ws = (unsigned char*)d_ws;
    size_t off = 0;
    float4* boxes_dec = (float4*)(ws + off); off += (size_t)BATCH * NANCH * sizeof(float4);
    int* cand_cnt     = (int*)(ws + off);    off += 256;
    unsigned long long* cand_keys = (unsigned long long*)(ws + off);
    off += (size_t)BATCH * CAND_MAX * sizeof(unsigned long long);
    float4* tk_box   = (float4*)(ws + off);  off += (size_t)BATCH * K_PRE * sizeof(float4);
    int* tk_label    = (int*)(ws + off);     off += (size_t)BATCH * K_PRE * sizeof(int);
    float* tk_score  = (float*)(ws + off);   off += (size_t)BATCH * K_PRE * sizeof(float);

    hipMemsetAsync(cand_cnt, 0, BATCH * sizeof(int), stream);

    int total = BATCH * NANCH;
    ssd_decode_softmax_kernel<<<(total + 255) / 256, 256, 0, stream>>>(
        bboxes, scores, dboxes, boxes_dec, cand_cnt, cand_keys);

    ssd_topk_sort_kernel<<<BATCH, 1024, (size_t)NPAD * 8, stream>>>(
        cand_keys, cand_cnt, boxes_dec, tk_box, tk_label, tk_score);

    size_t nms_smem = (size_t)K_PRE * NMS_WORDS * 4   // sup bit-matrix
                    + (size_t)K_PRE * 16              // boxes
                    + (size_t)K_PRE * 4               // labels
                    + (size_t)K_PRE * 4               // scores
                    + NMS_WORDS * 4;                  // keep words
    float* out_boxes  = (float*)d_out;
    int*   out_labels = (int*)((float*)d_out + (size_t)BATCH * MAXOUT * 4);
    float* out_scores = (float*)d_out + (size_t)BATCH * MAXOUT * 5;

    ssd_nms_out_kernel<<<BATCH, 1024, nms_smem, stream>>>(
        tk_box, tk_label, tk_score, out_boxes, out_labels, out_scores);
}